// GraphSAGE_31756988186712
// MI455X (gfx1250) — compile-verified
//
#include <hip/hip_runtime.h>

typedef __attribute__((ext_vector_type(2))) float v2f;
typedef __attribute__((ext_vector_type(8))) float v8f;

#define NN 50000
#define NE 800000
#define F  128
#define NG 64
#define NC 16
#define LDP 132   // LDS row stride (128 + 4 pad) -> conflict-free ds_load_b64

// ---------------------------------------------------------------- zero fill
__global__ void zero_f32(float* __restrict__ p, int n) {
    int i = blockIdx.x * blockDim.x + threadIdx.x;
    int stride = gridDim.x * blockDim.x;
    for (; i < n; i += stride) p[i] = 0.0f;
}

// ---------------------------------------------------------------- weight pair-pack
// Wt[p][n] = (W[2p][n], W[2p+1][n])  -> B fragments become one coalesced b64 load.
__global__ void pack_pairs(const float* __restrict__ W, float2* __restrict__ Wt,
                           int K, int N) {
    int i = blockIdx.x * blockDim.x + threadIdx.x;
    int total = (K >> 1) * N;
    if (i >= total) return;
    int p = i / N, n = i - p * N;
    Wt[i] = make_float2(W[(2 * p) * N + n], W[(2 * p + 1) * N + n]);
}

// ---------------------------------------------------------------- edge scatter
// One wave32 per edge: lane handles 4 consecutive floats of the 128-wide row.
// Node features (25.6MB) + accumulator live in L2; atomics resolve at L2.
__global__ void edge_scatter(const float* __restrict__ x,
                             const long long* __restrict__ src,
                             const long long* __restrict__ dst,
                             float* __restrict__ agg,
                             float* __restrict__ cnt) {
    const int wave = (blockIdx.x * blockDim.x + threadIdx.x) >> 5;
    const int lane = threadIdx.x & 31;
    if (wave >= NE) return;
    const int s = (int)src[wave];
    const int d = (int)dst[wave];
    const float4 v = *(const float4*)(x + (size_t)s * F + lane * 4);
    float* a = agg + (size_t)d * F + lane * 4;
    atomicAdd(a + 0, v.x);
    atomicAdd(a + 1, v.y);
    atomicAdd(a + 2, v.z);
    atomicAdd(a + 3, v.w);
    if (lane == 0) atomicAdd(cnt + d, 1.0f);
}

// ---------------------------------------------------------------- fused SAGE GEMM
// out = relu( (agg/max(cnt,1)) @ Wl + xin @ Wr + bl )
// Block = 8 waves = one tile_m (16 rows); waves cover the 8 column tiles.
// A staged in LDS (scale folded in); B from pair-packed weights: one b64/load.
__global__ void __launch_bounds__(256)
sage_gemm(const float* __restrict__ agg,
          const float* __restrict__ cnt,
          const float* __restrict__ xin,
          const v2f* __restrict__ WlT,   // (F/2) x F pairs
          const v2f* __restrict__ WrT,
          const float* __restrict__ bl,
          float* __restrict__ out) {
    __shared__ float sMean[16][LDP];
    __shared__ float sX[16][LDP];

    const int tile_m = blockIdx.x;             // 0..3124 (50000/16)
    const int tid    = threadIdx.x;

    // ---- cooperative stage: 16 rows x 32 float4 chunks per matrix ----
    for (int t = tid; t < 16 * 32; t += 256) {
        const int r = t >> 5;                  // 0..15
        const int c = (t & 31) << 2;           // 0,4,...,124
        const int row = tile_m * 16 + r;
        const float inv = 1.0f / fmaxf(cnt[row], 1.0f);
        const float4 va = *(const float4*)(agg + (size_t)row * F + c);
        sMean[r][c + 0] = va.x * inv;
        sMean[r][c + 1] = va.y * inv;
        sMean[r][c + 2] = va.z * inv;
        sMean[r][c + 3] = va.w * inv;
        const float4 vx = *(const float4*)(xin + (size_t)row * F + c);
        sX[r][c + 0] = vx.x;
        sX[r][c + 1] = vx.y;
        sX[r][c + 2] = vx.z;
        sX[r][c + 3] = vx.w;
    }
    __syncthreads();

    // ---- per-wave 16x16 tile: tile_n = wave id ----
    const int lane = tid & 31;
    const int m    = lane & 15;
    const int kgh  = lane >> 4;                // 0 or 1 (pair-row offset)
    const int col  = (tid >> 5) * 16 + m;      // global output column

    // per-lane B bases: fragment = pB[(k0>>1)*F] -> immediate-offset loads
    const v2f* pBl = WlT + (size_t)kgh * F + col;
    const v2f* pBr = WrT + (size_t)kgh * F + col;
    __builtin_prefetch(pBl, 0, 1);             // global_prefetch_b8
    __builtin_prefetch(pBr, 0, 1);

    v8f acc = {};
#pragma unroll 8
    for (int k0 = 0; k0 < F; k0 += 4) {
        const int kb = k0 + (kgh << 1);
        // mean @ Wl
        v2f a1 = *(const v2f*)(&sMean[m][kb]);             // ds_load_b64, bank-clean
        v2f b1 = pBl[(k0 >> 1) * F];                       // single coalesced b64
        acc = __builtin_amdgcn_wmma_f32_16x16x4_f32(false, a1, false, b1,
                                                    (short)0, acc, false, false);
        // x @ Wr
        v2f a2 = *(const v2f*)(&sX[m][kb]);
        v2f b2 = pBr[(k0 >> 1) * F];
        acc = __builtin_amdgcn_wmma_f32_16x16x4_f32(false, a2, false, b2,
                                                    (short)0, acc, false, false);
    }

    const float bias = bl[col];
#pragma unroll
    for (int r = 0; r < 8; ++r) {
        const int rr = (lane < 16) ? r : (r + 8);  // C/D layout: VGPR r -> M=r / M=r+8
        const float v = acc[r] + bias;
        out[(size_t)(tile_m * 16 + rr) * F + col] = fmaxf(v, 0.0f);
    }
}

// ---------------------------------------------------------------- graph pooling
__global__ void pool_scatter(const float* __restrict__ h,
                             const long long* __restrict__ batch,
                             float* __restrict__ gsum,
                             float* __restrict__ gcnt) {
    const int wave = (blockIdx.x * blockDim.x + threadIdx.x) >> 5;
    const int lane = threadIdx.x & 31;
    if (wave >= NN) return;
    const int b = (int)batch[wave];
    const float4 v = *(const float4*)(h + (size_t)wave * F + lane * 4);
    float* g = gsum + b * F + lane * 4;
    atomicAdd(g + 0, v.x);
    atomicAdd(g + 1, v.y);
    atomicAdd(g + 2, v.z);
    atomicAdd(g + 3, v.w);
    if (lane == 0) atomicAdd(gcnt + b, 1.0f);
}

// ---------------------------------------------------------------- FC1: emb = (gsum/cnt) @ Wfc1 + bfc1   (64x128 @ 128x128)
__global__ void fc1_gemm(const float* __restrict__ gsum,
                         const float* __restrict__ gcnt,
                         const v2f* __restrict__ WT,
                         const float* __restrict__ b,
                         float* __restrict__ emb) {
    const int wave   = threadIdx.x >> 5;       // 0..31
    const int lane   = threadIdx.x & 31;
    const int tile_m = wave >> 3;              // 0..3
    const int tile_n = wave & 7;               // 0..7
    const int m   = lane & 15;
    const int kgh = lane >> 4;
    const int row = tile_m * 16 + m;
    const int col = tile_n * 16 + m;

    const float inv = 1.0f / fmaxf(gcnt[row], 1.0f);
    const float* arow = gsum + (size_t)row * F;
    const v2f* pB = WT + (size_t)kgh * F + col;

    v8f acc = {};
#pragma unroll 8
    for (int k0 = 0; k0 < F; k0 += 4) {
        const int kb = k0 + (kgh << 1);
        v2f a = *(const v2f*)(arow + kb);
        a *= inv;
        v2f bb = pB[(k0 >> 1) * F];
        acc = __builtin_amdgcn_wmma_f32_16x16x4_f32(false, a, false, bb,
                                                    (short)0, acc, false, false);
    }
    const float bias = b[col];
#pragma unroll
    for (int r = 0; r < 8; ++r) {
        const int rr = (lane < 16) ? r : (r + 8);
        emb[(size_t)(tile_m * 16 + rr) * F + col] = acc[r] + bias;
    }
}

// ---------------------------------------------------------------- FC2: out = emb @ Wfc2 + bfc2   (64x128 @ 128x16)
__global__ void fc2_gemm(const float* __restrict__ emb,
                         const v2f* __restrict__ WT,
                         const float* __restrict__ b,
                         float* __restrict__ out) {
    const int wave = threadIdx.x >> 5;         // tile_m 0..3
    const int lane = threadIdx.x & 31;
    const int m   = lane & 15;
    const int kgh = lane >> 4;
    const int row = wave * 16 + m;
    const int col = m;                          // single N tile (NC==16)

    const float* arow = emb + (size_t)row * F;
    const v2f* pB = WT + (size_t)kgh * NC + col;

    v8f acc = {};
#pragma unroll 8
    for (int k0 = 0; k0 < F; k0 += 4) {
        const int kb = k0 + (kgh << 1);
        v2f a  = *(const v2f*)(arow + kb);
        v2f bb = pB[(k0 >> 1) * NC];
        acc = __builtin_amdgcn_wmma_f32_16x16x4_f32(false, a, false, bb,
                                                    (short)0, acc, false, false);
    }
    const float bias = b[col];
#pragma unroll
    for (int r = 0; r < 8; ++r) {
        const int rr = (lane < 16) ? r : (r + 8);
        out[(size_t)(wave * 16 + rr) * NC + col] = acc[r] + bias;
    }
}

// ---------------------------------------------------------------- host launch
extern "C" void kernel_launch(void* const* d_in, const int* in_sizes, int n_in,
                              void* d_out, int out_size, void* d_ws, size_t ws_size,
                              hipStream_t stream) {
    const float*     x     = (const float*)d_in[0];
    const long long* ei    = (const long long*)d_in[1];
    const long long* batch = (const long long*)d_in[2];
    const float* Wl[4] = { (const float*)d_in[3], (const float*)d_in[6],
                           (const float*)d_in[9], (const float*)d_in[12] };
    const float* bl[4] = { (const float*)d_in[4], (const float*)d_in[7],
                           (const float*)d_in[10], (const float*)d_in[13] };
    const float* Wr[4] = { (const float*)d_in[5], (const float*)d_in[8],
                           (const float*)d_in[11], (const float*)d_in[14] };
    const float* Wfc1 = (const float*)d_in[15];
    const float* bfc1 = (const float*)d_in[16];
    const float* Wfc2 = (const float*)d_in[17];
    const float* bfc2 = (const float*)d_in[18];

    // Workspace layout (floats):
    // agg | cnt | hA | hB | gsum | gcnt | emb | WlT[4] WrT[4] | WT1 | WT2
    float* ws   = (float*)d_ws;
    float* agg  = ws;
    float* cnt  = agg + (size_t)NN * F;
    float* hA   = cnt + NN;
    float* hB   = hA + (size_t)NN * F;
    float* gsum = hB + (size_t)NN * F;
    float* gcnt = gsum + NG * F;
    float* emb  = gcnt + NG;
    float* wptr = emb + NG * F;
    const int WPAIR = (F / 2) * F * 2;         // floats per packed 128x128 matrix
    float* WlT[4]; float* WrT[4];
    for (int l = 0; l < 4; ++l) { WlT[l] = wptr; wptr += WPAIR; }
    for (int l = 0; l < 4; ++l) { WrT[l] = wptr; wptr += WPAIR; }
    float* WT1 = wptr; wptr += WPAIR;
    float* WT2 = wptr; wptr += (F / 2) * NC * 2;

    const long long* src = ei;
    const long long* dst = ei + NE;

    // ---- pack all weights into pair-interleaved layout (once per launch) ----
    const int PPK = ((F / 2) * F + 255) / 256; // 32 blocks
    for (int l = 0; l < 4; ++l) {
        pack_pairs<<<PPK, 256, 0, stream>>>(Wl[l], (float2*)WlT[l], F, F);
        pack_pairs<<<PPK, 256, 0, stream>>>(Wr[l], (float2*)WrT[l], F, F);
    }
    pack_pairs<<<PPK, 256, 0, stream>>>(Wfc1, (float2*)WT1, F, F);
    pack_pairs<<<4, 256, 0, stream>>>(Wfc2, (float2*)WT2, F, NC);

    const int ZB = 2048;                 // zero-fill blocks (grid-stride)
    const int scatter_blocks = NE / 8;   // 8 waves (edges) per 256-thr block
    const int gemm_blocks = NN / 16;     // one block per 16-row tile

    const float* hin = x;
    float* bufs[2] = { hA, hB };
    for (int l = 0; l < 4; ++l) {
        zero_f32<<<ZB, 256, 0, stream>>>(agg, NN * F + NN);  // agg & cnt contiguous
        edge_scatter<<<scatter_blocks, 256, 0, stream>>>(hin, src, dst, agg, cnt);
        float* hout = bufs[l & 1];
        sage_gemm<<<gemm_blocks, 256, 0, stream>>>(agg, cnt, hin,
                                                   (const v2f*)WlT[l],
                                                   (const v2f*)WrT[l],
                                                   bl[l], hout);
        hin = hout;
    }

    zero_f32<<<64, 256, 0, stream>>>(gsum, NG * F + NG);     // gsum & gcnt contiguous
    pool_scatter<<<(NN + 7) / 8, 256, 0, stream>>>(hin, batch, gsum, gcnt);
    fc1_gemm<<<1, 1024, 0, stream>>>(gsum, gcnt, (const v2f*)WT1, bfc1, emb);
    fc2_gemm<<<1, 128, 0, stream>>>(emb, (const v2f*)WT2, bfc2, (float*)d_out);
}